// DePatchEmbed_21552145891783
// MI455X (gfx1250) — compile-verified
//
#include <hip/hip_runtime.h>
#include <cstdint>

// DePatchEmbed permutation:
//   out[n, c, i*8+a, j*8+b] = x[n, j*32+i, c*64 + a*8 + b]
// N=4, C=256, H=W=256, p=8, Gi=Gj=32.
// Pure copy: 256 MiB in + 256 MiB out -> HBM roofline ~22.5 us @ 23.3 TB/s.
// One workgroup per (n,c,i): reads 32 x 256B chunks (strided 2 MiB, each
// 2 full cachelines), writes one contiguous 8 KB block. Permutation done in
// LDS, staged via CDNA5 async global->LDS loads (ASYNCcnt).

typedef float v4f __attribute__((ext_vector_type(4)));
typedef int   v4i __attribute__((ext_vector_type(4)));

#ifndef __has_builtin
#define __has_builtin(x) 0
#endif

#if defined(__AMDGCN__) && \
    __has_builtin(__builtin_amdgcn_global_load_async_to_lds_b128) && \
    __has_builtin(__builtin_amdgcn_s_wait_asynccnt)
#define DEPATCH_ASYNC 1
#else
#define DEPATCH_ASYNC 0
#endif

namespace {
constexpr int kN   = 4;
constexpr int kC   = 256;
constexpr int kP   = 8;
constexpr int kGi  = 32;                 // row-block index (fastest patch dim)
constexpr int kGj  = 32;                 // col-block index
constexpr int kDim = kC * kP * kP;       // 16384 floats per patch
constexpr long kJStride = (long)kGi * kDim;  // 524288 floats between j-chunks
constexpr int kTileF4 = (kP * 256) / 4;  // 512 float4 = 8 KB per (n,c,i) tile
constexpr int kLdsStride = 17;           // 16 float4 per j-chunk + 1 pad
}

__global__ __launch_bounds__(256) void depatch_embed_kernel(
    const float* __restrict__ x, float* __restrict__ out) {
  const int bid = blockIdx.x;            // 32768 blocks = N*C*Gi
  const int i = bid & (kGi - 1);
  const int c = (bid >> 5) & (kC - 1);
  const int n = bid >> 13;
  const int t = threadIdx.x;             // 256 threads = 8 waves

  // Input chunk base: x[n][j*32+i][c*64], j-chunks are 64 contiguous floats.
  const float* __restrict__ src =
      x + ((long)n * (kGi * kGj) + i) * kDim + (long)c * (kP * kP);
  // Output tile base: out[n][c][i*8][0], 8 rows x 256 cols contiguous (8 KB).
  float* __restrict__ dst =
      out + (((long)n * kC + c) * 256 + (long)i * kP) * 256;

#if DEPATCH_ASYNC
  __shared__ v4f tile[kGj * kLdsStride];  // ~8.7 KB, bank-conflict padded

  // Stage 1: async DMA global -> LDS in input-linear order.
  // f = j*16 + r : 16 consecutive lanes fetch one 256B chunk (B128/lane).
  #pragma unroll
  for (int f = t; f < kTileF4; f += 256) {
    const int j = f >> 4;
    const int r = f & 15;
    __builtin_amdgcn_global_load_async_to_lds_b128(
        (__attribute__((address_space(1))) v4i*)(src + (long)j * kJStride + r * 4),
        (__attribute__((address_space(3))) v4i*)(&tile[j * kLdsStride + r]),
        0, /*cpol=*/0);
  }
  __builtin_amdgcn_s_wait_asynccnt(0);
  __syncthreads();

  // Stage 2: permuted LDS read (2-way conflicts max thanks to stride-17),
  // fully coalesced streaming store of the contiguous 8 KB output tile.
  #pragma unroll
  for (int g = t; g < kTileF4; g += 256) {
    const int a  = g >> 6;         // 0..7   (row within patch)
    const int j  = (g >> 1) & 31;  // 0..31  (col block)
    const int bq = g & 1;          // float4 half within 8-float row segment
    v4f v = tile[j * kLdsStride + a * 2 + bq];
    __builtin_nontemporal_store(v, (v4f*)(dst + g * 4));
  }
#else
  // Fallback: direct copy, output-coalesced; input reads are 32B-granular but
  // every 128B line is fully consumed inside this workgroup (a=0..7 rows).
  #pragma unroll
  for (int g = t; g < kTileF4; g += 256) {
    const int a  = g >> 6;
    const int j  = (g >> 1) & 31;
    const int bq = g & 1;
    v4f v = __builtin_nontemporal_load(
        (const v4f*)(src + (long)j * kJStride + a * kP + bq * 4));
    __builtin_nontemporal_store(v, (v4f*)(dst + g * 4));
  }
#endif
}

extern "C" void kernel_launch(void* const* d_in, const int* in_sizes, int n_in,
                              void* d_out, int out_size, void* d_ws, size_t ws_size,
                              hipStream_t stream) {
  const float* x = (const float*)d_in[0];   // (4, 1024, 16384) fp32
  float* out = (float*)d_out;               // (4, 256, 256, 256) fp32
  (void)in_sizes; (void)n_in; (void)out_size; (void)d_ws; (void)ws_size;

  const int blocks = kN * kC * kGi;         // 32768
  depatch_embed_kernel<<<dim3(blocks), dim3(256), 0, stream>>>(x, out);
}